// EdgeFeatGAE_23630910062818
// MI455X (gfx1250) — compile-verified
//
#include <hip/hip_runtime.h>
#include <hip/hip_bf16.h>

typedef float v2f __attribute__((ext_vector_type(2)));
typedef float v8f __attribute__((ext_vector_type(8)));

#define FEAT 128
#define HID  32
#define EMB  16

// ---------------------------------------------------------------------------
// H0 = X @ W1   (X: [N,128] row-major, W1: [128,32] row-major, H0: [N,32])
// One wave32 per 16x16 output tile, V_WMMA_F32_16X16X4_F32, 32 steps over K.
// ---------------------------------------------------------------------------
__global__ void gemm_xw1_wmma(const float* __restrict__ X,
                              const float* __restrict__ W1,
                              float* __restrict__ H0, int nRows) {
    const int wave  = (blockIdx.x * blockDim.x + threadIdx.x) >> 5;
    const int lane  = threadIdx.x & 31;
    const int nTile = wave & 1;          // 32 cols -> 2 N-tiles
    const int mTile = wave >> 1;
    if (mTile * 16 >= nRows) return;     // wave-uniform exit (EXEC stays all-1)

    const int m0   = mTile * 16;
    const int n0   = nTile * 16;
    const int half = lane >> 4;          // 0: K pair {0,1}, 1: K pair {2,3}
    const int l    = lane & 15;          // row (A) / col (B) within tile

    const float* __restrict__ xrow = X + (size_t)(m0 + l) * FEAT;

    v8f c = {};
    #pragma unroll 8
    for (int k0 = 0; k0 < FEAT; k0 += 4) {
        v2f a, b;
        const int ka = k0 + 2 * half;
        // A fragment: 16x4 f32, row = l, K = ka, ka+1
        a.x = xrow[ka];
        a.y = xrow[ka + 1];
        // B fragment: 4x16 f32, col = n0+l, K = ka, ka+1
        b.x = W1[(size_t)ka       * HID + n0 + l];
        b.y = W1[(size_t)(ka + 1) * HID + n0 + l];
        c = __builtin_amdgcn_wmma_f32_16x16x4_f32(
                /*neg_a=*/false, a, /*neg_b=*/false, b,
                /*c_mod=*/(short)0, c, /*reuse_a=*/false, /*reuse_b=*/false);
    }

    // C/D layout: VGPR i -> row m0+i (lanes 0-15) / m0+8+i (lanes 16-31)
    const int col   = n0 + l;
    const int rbase = m0 + half * 8;
    #pragma unroll
    for (int i = 0; i < 8; ++i)
        H0[(size_t)(rbase + i) * HID + col] = c[i];
}

// ---------------------------------------------------------------------------
// tiny reductions of the second-layer weights (output sums over embed dim)
// w2s[j] = sum_k W2[j,k];  sumb2 = sum_k b2[k]
// ---------------------------------------------------------------------------
__global__ void reduce_w2_kernel(const float* __restrict__ W2,
                                 const float* __restrict__ b2,
                                 float* __restrict__ w2s,
                                 float* __restrict__ sumb2) {
    int j = threadIdx.x;
    if (j < HID) {
        float s = 0.f;
        #pragma unroll
        for (int k = 0; k < EMB; ++k) s += W2[j * EMB + k];
        w2s[j] = s;
    }
    if (j == 0) {
        float s = 0.f;
        #pragma unroll
        for (int k = 0; k < EMB; ++k) s += b2[k];
        sumb2[0] = s;
    }
}

// ---------------------------------------------------------------------------
// per-relation pipeline
// ---------------------------------------------------------------------------
__global__ void init_deg_kernel(float* __restrict__ deg, int N) {
    int n = blockIdx.x * blockDim.x + threadIdx.x;
    if (n < N) deg[n] = 1.0f;            // self loop
}

__global__ void count_deg_kernel(float* __restrict__ deg,
                                 const int* __restrict__ dst, int E) {
    int e = blockIdx.x * blockDim.x + threadIdx.x;
    if (e < E) atomicAdd(&deg[dst[e]], 1.0f);
}

__global__ void rsqrt_inplace_kernel(float* __restrict__ deg, int N) {
    int n = blockIdx.x * blockDim.x + threadIdx.x;
    if (n < N) deg[n] = rsqrtf(deg[n]);  // deg >= 1 always (self loop)
}

// agg[n,:] = dinv[n]^2 * H0[n,:]   (the self-loop message of layer 1)
__global__ void init_agg_kernel(float* __restrict__ agg,
                                const float* __restrict__ H0,
                                const float* __restrict__ dinv, int N) {
    int tid = blockIdx.x * blockDim.x + threadIdx.x;
    int n = tid >> 5;
    if (n >= N) return;
    float di = dinv[n];
    agg[tid] = di * di * H0[tid];
}

// wave per edge, lane = feature: agg[dst,:] += dinv[s]*dinv[d] * H0[src,:]
__global__ void scatter_l1_kernel(float* __restrict__ agg,
                                  const float* __restrict__ H0,
                                  const float* __restrict__ dinv,
                                  const int* __restrict__ src,
                                  const int* __restrict__ dst, int E) {
    int tid = blockIdx.x * blockDim.x + threadIdx.x;
    int e = tid >> 5;
    if (e >= E) return;
    int lane = tid & 31;
    int s = src[e], d = dst[e];
    float norm = dinv[s] * dinv[d];
    atomicAdd(&agg[(size_t)d * HID + lane], norm * H0[(size_t)s * HID + lane]);
}

// h = relu(agg + b1); sval[n] = h . w2s; outrow[n] = dinv[n]^2*sval[n] + sumb2
__global__ void node_post_kernel(const float* __restrict__ agg,
                                 const float* __restrict__ b1,
                                 const float* __restrict__ w2s,
                                 const float* __restrict__ dinv,
                                 const float* __restrict__ sumb2,
                                 float* __restrict__ sval,
                                 float* __restrict__ outrow, int N) {
    int tid = blockIdx.x * blockDim.x + threadIdx.x;
    int n = tid >> 5;
    if (n >= N) return;
    int j = tid & 31;
    float h = agg[(size_t)n * HID + j] + b1[j];
    h = h > 0.f ? h : 0.f;
    float p = h * w2s[j];
    // wave32 butterfly reduction
    #pragma unroll
    for (int off = 16; off > 0; off >>= 1)
        p += __shfl_xor(p, off, 32);
    if (j == 0) {
        sval[n] = p;
        float di = dinv[n];
        outrow[n] = di * di * p + sumb2[0];   // self-loop term of layer 2
    }
}

// outrow[dst] += dinv[s]*dinv[d] * sval[src]
__global__ void scatter_l2_kernel(float* __restrict__ outrow,
                                  const float* __restrict__ sval,
                                  const float* __restrict__ dinv,
                                  const int* __restrict__ src,
                                  const int* __restrict__ dst, int E) {
    int e = blockIdx.x * blockDim.x + threadIdx.x;
    if (e >= E) return;
    int s = src[e], d = dst[e];
    atomicAdd(&outrow[d], dinv[s] * dinv[d] * sval[s]);
}

// ---------------------------------------------------------------------------
extern "C" void kernel_launch(void* const* d_in, const int* in_sizes, int n_in,
                              void* d_out, int out_size, void* d_ws, size_t ws_size,
                              hipStream_t stream) {
    const float* x  = (const float*)d_in[0];
    const int*   eis = (const int*)d_in[1];
    const float* W1 = (const float*)d_in[2];
    const float* b1 = (const float*)d_in[3];
    const float* W2 = (const float*)d_in[4];
    const float* b2 = (const float*)d_in[5];
    float* out = (float*)d_out;

    const int N = in_sizes[0] / FEAT;            // 100000
    const int R = 4;
    const int E = in_sizes[1] / (2 * R);         // 1600000

    // workspace partition (floats)
    float* H0    = (float*)d_ws;                 // N*32
    float* agg   = H0  + (size_t)N * HID;        // N*32
    float* dinv  = agg + (size_t)N * HID;        // N   (deg -> dinv in place)
    float* sval  = dinv + N;                     // N
    float* w2s   = sval + N;                     // 32
    float* sumb2 = w2s + HID;                    // 1
    (void)ws_size; (void)n_in; (void)out_size;

    // shared across relations: GEMM + weight reductions
    {
        int waves  = (N / 16) * (HID / 16);      // 12500
        int blocks = (waves + 7) / 8;            // 8 waves / 256-thread block
        gemm_xw1_wmma<<<blocks, 256, 0, stream>>>(x, W1, H0, N);
        reduce_w2_kernel<<<1, 32, 0, stream>>>(W2, b2, w2s, sumb2);
    }

    const int TB = 256;
    const int gN   = (N + TB - 1) / TB;
    const int gE   = (E + TB - 1) / TB;
    const int gN32 = ((N * HID) + TB - 1) / TB;
    const int gE32 = (int)(((size_t)E * 32 + TB - 1) / TB);

    for (int r = 0; r < R; ++r) {
        const int* src = eis + (size_t)r * 2 * E;
        const int* dst = src + E;
        float* outrow = out + (size_t)r * N;

        init_deg_kernel<<<gN, TB, 0, stream>>>(dinv, N);
        count_deg_kernel<<<gE, TB, 0, stream>>>(dinv, dst, E);
        rsqrt_inplace_kernel<<<gN, TB, 0, stream>>>(dinv, N);

        init_agg_kernel<<<gN32, TB, 0, stream>>>(agg, H0, dinv, N);
        scatter_l1_kernel<<<gE32, TB, 0, stream>>>(agg, H0, dinv, src, dst, E);

        node_post_kernel<<<gN32, TB, 0, stream>>>(agg, b1, w2s, dinv, sumb2,
                                                  sval, outrow, N);
        scatter_l2_kernel<<<gE, TB, 0, stream>>>(outrow, sval, dinv, src, dst, E);
    }
}